// GCNLayer_7988639171161
// MI455X (gfx1250) — compile-verified
//
#include <hip/hip_runtime.h>
#include <hip/hip_bf16.h>

typedef __attribute__((ext_vector_type(2))) float v2f;
typedef __attribute__((ext_vector_type(8))) float v8f;

#define D 48   // feature dim (D_IN == D_OUT == 48)

// ---------------------------------------------------------------------------
// Phase 1: degree histograms via native f32 L2 atomics
// ---------------------------------------------------------------------------
__global__ __launch_bounds__(256) void gcn_degree_kernel(
    const int* __restrict__ src, const int* __restrict__ dst,
    float* __restrict__ out_deg, float* __restrict__ in_deg, int n_edges)
{
    int e = blockIdx.x * blockDim.x + threadIdx.x;
    if (e < n_edges) {
        unsafeAtomicAdd(&out_deg[src[e]], 1.0f);
        unsafeAtomicAdd(&in_deg[dst[e]], 1.0f);
    }
}

// ---------------------------------------------------------------------------
// Phase 2: msg = features[src] * rsqrt(max(out_deg[src],1)); scatter-sum to agg[dst]
// 192-thread blocks = 6 wave32 = 4 edges/block; 48 consecutive lanes per edge
// -> coalesced feature gathers, contiguous atomic bursts into one agg row.
// ---------------------------------------------------------------------------
__global__ __launch_bounds__(192) void gcn_scatter_kernel(
    const float* __restrict__ features,
    const int* __restrict__ src, const int* __restrict__ dst,
    const float* __restrict__ out_deg, float* __restrict__ agg, int n_edges)
{
    int t = blockIdx.x * 192 + threadIdx.x;
    int e = t / D;
    int f = t - e * D;
    if (e < n_edges) {
        int s = src[e];
        float ci = rsqrtf(fmaxf(out_deg[s], 1.0f));
        float v  = features[(size_t)s * D + f] * ci;
        unsafeAtomicAdd(&agg[(size_t)dst[e] * D + f], v);
    }
}

// ---------------------------------------------------------------------------
// Phase 3: out = relu(h @ W^T + b), h = (in_deg>0) ? agg*rsqrt(in_deg) : features
// One wave32 computes one 16x16 output tile with V_WMMA_F32_16X16X4_F32.
// K = 48 -> 12 wmma steps of K=4. 50000 rows = 3125 exact M-tiles, 48 = 3 N-tiles.
//
// A(16x4) lane layout: lanes 0-15 -> M=lane, K in {k0, k0+1}
//                      lanes 16-31 -> M=lane-16, K in {k0+2, k0+3}
// B(4x16) lane layout: mirrored; B[k][n] = W[n*48 + k]
// C/D(16x16): VGPR r -> M = r + (lane>=16 ? 8 : 0), N = lane & 15
// ---------------------------------------------------------------------------
__global__ __launch_bounds__(256) void gcn_gemm_wmma_kernel(
    const float* __restrict__ features, const float* __restrict__ agg,
    const float* __restrict__ in_deg,  const float* __restrict__ W,
    const float* __restrict__ bias,    float* __restrict__ out, int n_nodes)
{
    const int lane = threadIdx.x & 31;
    const int wave = threadIdx.x >> 5;

    const int n_tiles_m = (n_nodes + 15) >> 4;
    const int n_tiles   = n_tiles_m * (D / 16);
    const int tile      = blockIdx.x * 8 + wave;
    if (tile >= n_tiles) return;   // wave-uniform: EXEC stays all-1s for WMMA

    const int tm = tile / (D / 16);
    const int tn = tile - tm * (D / 16);

    // Per-lane fixed A-row for the whole tile
    const int  mrow  = lane & 15;
    int        row   = tm * 16 + mrow;
    const int  rowc  = row < n_nodes ? row : n_nodes - 1;   // clamp, keep EXEC full
    const int  khalf = (lane >> 4) << 1;                    // 0 or 2

    const float indeg = in_deg[rowc];
    float scale;
    const float* __restrict__ arow;
    if (indeg > 0.0f) {
        scale = rsqrtf(indeg);                              // cj
        arow  = agg + (size_t)rowc * D;
    } else {
        scale = 1.0f;                                       // zero in-degree: keep h
        arow  = features + (size_t)rowc * D;
    }

    // Per-lane fixed B-column (output column)
    const int ncol = tn * 16 + (lane & 15);
    const float* __restrict__ wrow = W + (size_t)ncol * D;  // B[k][ncol] = W[ncol*D + k]

    v8f c = {};
#pragma unroll
    for (int kk = 0; kk < D / 4; ++kk) {
        const int k0 = kk * 4 + khalf;
        v2f a, b;
        a.x = arow[k0]     * scale;
        a.y = arow[k0 + 1] * scale;
        b.x = wrow[k0];
        b.y = wrow[k0 + 1];
        c = __builtin_amdgcn_wmma_f32_16x16x4_f32(
                /*neg_a=*/false, a, /*neg_b=*/false, b,
                /*c_mod=*/(short)0, c, /*reuse_a=*/false, /*reuse_b=*/false);
    }

    // Epilogue: bias + ReLU, scatter 8 accumulator rows
    const float bv    = bias[ncol];
    const int   rbase = (lane >> 4) << 3;                   // 0 or 8
#pragma unroll
    for (int r = 0; r < 8; ++r) {
        const int orow = tm * 16 + rbase + r;
        if (orow < n_nodes) {
            float v = c[r] + bv;
            out[(size_t)orow * D + ncol] = v > 0.0f ? v : 0.0f;
        }
    }
}

// ---------------------------------------------------------------------------
// Host-side launcher (graph-capture safe: only async ops on `stream`)
// ---------------------------------------------------------------------------
extern "C" void kernel_launch(void* const* d_in, const int* in_sizes, int n_in,
                              void* d_out, int out_size, void* d_ws, size_t ws_size,
                              hipStream_t stream) {
    const float* features = (const float*)d_in[0];
    const int*   src      = (const int*)  d_in[1];
    const int*   dst      = (const int*)  d_in[2];
    const float* W        = (const float*)d_in[3];
    const float* bias     = (const float*)d_in[4];
    float*       out      = (float*)d_out;

    const int n_nodes = in_sizes[0] / D;
    const int n_edges = in_sizes[1];

    // workspace: out_deg[n] | in_deg[n] | agg[n*D]
    float* out_deg = (float*)d_ws;
    float* in_deg  = out_deg + n_nodes;
    float* agg     = in_deg + n_nodes;
    const size_t zero_bytes = (size_t)(2 * n_nodes + (size_t)n_nodes * D) * sizeof(float);
    hipMemsetAsync(d_ws, 0, zero_bytes, stream);

    // Phase 1: degrees
    {
        int blocks = (n_edges + 255) / 256;
        gcn_degree_kernel<<<blocks, 256, 0, stream>>>(src, dst, out_deg, in_deg, n_edges);
    }
    // Phase 2: gather-scale-scatter
    {
        long long total = (long long)n_edges * D;
        int blocks = (int)((total + 191) / 192);
        gcn_scatter_kernel<<<blocks, 192, 0, stream>>>(features, src, dst, out_deg, agg, n_edges);
    }
    // Phase 3: WMMA GEMM + bias + ReLU
    {
        int n_tiles = ((n_nodes + 15) / 16) * (D / 16);
        int blocks  = (n_tiles + 7) / 8;   // 8 waves (tiles) per 256-thread block
        gcn_gemm_wmma_kernel<<<blocks, 256, 0, stream>>>(features, agg, in_deg, W, bias, out, n_nodes);
    }
}